// DOMPooling_60361470378072
// MI455X (gfx1250) — compile-verified
//
#include <hip/hip_runtime.h>
#include <math.h>
#include <float.h>
#include <stdint.h>

typedef float f2  __attribute__((ext_vector_type(2)));
typedef float f4  __attribute__((ext_vector_type(4)));
typedef float v8f __attribute__((ext_vector_type(8)));

#define EMBED   128
#define PROJ_IN 260   // 2*EMBED + 4

// Native CDNA5 float atomic max (no return -> STOREcnt), device scope so the
// RMW resolves in L2 where the accumulators live.
__device__ __forceinline__ void atomic_max_f32_dev(float* p, float v) {
  asm volatile("global_atomic_max_num_f32 %0, %1, off scope:SCOPE_DEV"
               :: "v"((unsigned long long)(uintptr_t)p), "v"(v)
               : "memory");
}

// ---------------- kernel 1: init workspace ----------------
__global__ void init_ws(float* __restrict__ sum, float* __restrict__ mx,
                        float* __restrict__ cnt, int n_elems, int n_doms) {
  int i  = blockIdx.x * blockDim.x + threadIdx.x;
  int i4 = i * 4;
  if (i4 < n_elems) {
    f4 z = {0.f, 0.f, 0.f, 0.f};
    f4 m = {-FLT_MAX, -FLT_MAX, -FLT_MAX, -FLT_MAX};
    *(f4*)(sum + i4) = z;
    *(f4*)(mx  + i4) = m;
  }
  if (i < n_doms) cnt[i] = 0.f;
}

// ---------------- kernel 2: scatter segment sum/max ----------------
// One wave32 per pulse row; lane L owns floats [4L, 4L+4) of the 128-wide row.
__global__ void __launch_bounds__(256)
scatter_pool(const float* __restrict__ emb, const int* __restrict__ p2d,
             float* __restrict__ sum, float* __restrict__ mx,
             float* __restrict__ cnt, int n_pulses) {
  int wave = (int)((blockIdx.x * blockDim.x + threadIdx.x) >> 5);
  int lane = threadIdx.x & 31;
  if (wave >= n_pulses) return;                 // wave-uniform guard
  int dom = p2d[wave];
  const f4* row = (const f4*)(emb + (size_t)wave * EMBED);
  f4 v = __builtin_nontemporal_load(row + lane);  // NT: don't evict L2 accums
  float* s = sum + (size_t)dom * EMBED + lane * 4;
  float* m = mx  + (size_t)dom * EMBED + lane * 4;
  atomicAdd(s + 0, v.x); atomicAdd(s + 1, v.y);
  atomicAdd(s + 2, v.z); atomicAdd(s + 3, v.w);
  atomic_max_f32_dev(m + 0, v.x); atomic_max_f32_dev(m + 1, v.y);
  atomic_max_f32_dev(m + 2, v.z); atomic_max_f32_dev(m + 3, v.w);
  if (lane == 0) atomicAdd(cnt + dom, 1.0f);
}

// ---------------- kernel 3: finalize feats + f32 WMMA GEMM ----------------
// Block = 256 threads (8 waves) handles 16 DOM rows x all 128 output columns.
// Wave w produces the 16x16 output tile at columns [16w, 16w+16).
__global__ void __launch_bounds__(256)
pool_gemm(const float* __restrict__ sum, const float* __restrict__ mx,
          const float* __restrict__ cnt,
          const float* __restrict__ dom_min_time,
          const float* __restrict__ dom_sum_charge,
          const float* __restrict__ dom_std_time,
          const int*   __restrict__ dom_pulse_counts,
          const float* __restrict__ W, const float* __restrict__ bvec,
          float* __restrict__ out, int n_doms) {
  __shared__ float s_feats[16][PROJ_IN];   // 16.6 KB
  __shared__ float s_cnt[16];

  int tid    = threadIdx.x;
  int block0 = blockIdx.x * 16;

  if (tid < 16) {
    int dom = block0 + tid;
    s_cnt[tid] = (dom < n_doms) ? cnt[dom] : 0.f;
  }
  __syncthreads();

  // Build the 16x260 feats tile cooperatively.
  for (int e = tid; e < 16 * PROJ_IN; e += 256) {
    int r = e / PROJ_IN;
    int c = e - r * PROJ_IN;
    int dom = block0 + r;
    float val = 0.f;
    if (dom < n_doms) {
      float cn = s_cnt[r];
      if (c < EMBED) {                       // mean pool
        val = sum[(size_t)dom * EMBED + c] / fmaxf(cn, 1.0f);
      } else if (c < 2 * EMBED) {            // max pool (empty segment -> 0)
        float mv = mx[(size_t)dom * EMBED + (c - EMBED)];
        val = (cn > 0.f) ? mv : 0.f;
      } else if (c == 2 * EMBED) {
        val = (dom_min_time[dom] - 10000.0f) * (1.0f / 30000.0f);
      } else if (c == 2 * EMBED + 1) {
        val = log10f(fmaxf(dom_sum_charge[dom], 1e-6f)) * (1.0f / 3.0f);
      } else if (c == 2 * EMBED + 2) {
        val = dom_std_time[dom] * (1.0f / 30000.0f);
      } else {
        val = log1pf((float)dom_pulse_counts[dom]) * (1.0f / 3.0f) - 1.0f;
      }
    }
    s_feats[r][c] = val;
  }
  __syncthreads();

  int waveId = tid >> 5;
  int lane   = tid & 31;
  int m      = lane & 15;                    // A: lane -> M
  int khalf  = (lane >> 4) << 1;             // upper half-wave owns K+2,K+3
  int n      = (waveId << 4) + m;            // B/D: lane -> N

  v8f acc = {};
  #pragma unroll 5
  for (int k0 = 0; k0 < PROJ_IN; k0 += 4) {  // 65 K-steps, exact
    int kk = k0 + khalf;
    f2 a = *(const f2*)&s_feats[m][kk];      // ds_load_2addr_b64, conflict-free
    f2 b;
    b.x = W[(size_t)kk * EMBED + n];
    b.y = W[(size_t)(kk + 1) * EMBED + n];
    acc = __builtin_amdgcn_wmma_f32_16x16x4_f32(
        /*neg_a=*/false, a, /*neg_b=*/false, b,
        /*c_mod=*/(short)0, acc, /*reuse_a=*/false, /*reuse_b=*/false);
  }

  float bias = bvec[n];
  int mbase = (lane >> 4) << 3;              // D: VGPR r -> M = r + 8*(lane>=16)
  float* orow = out + (size_t)(block0 + mbase) * EMBED + n;

  if (block0 + 16 <= n_doms) {
    // Block-uniform fast path: 8 unguarded NT stores off one base address,
    // immediate offsets r*512B -> single store clause, no exec juggling.
    #pragma unroll
    for (int r = 0; r < 8; ++r)
      __builtin_nontemporal_store(acc[r] + bias, orow + r * EMBED);
  } else {
    #pragma unroll
    for (int r = 0; r < 8; ++r) {
      int dom = block0 + mbase + r;
      if (dom < n_doms)
        out[(size_t)dom * EMBED + n] = acc[r] + bias;
    }
  }
}

extern "C" void kernel_launch(void* const* d_in, const int* in_sizes, int n_in,
                              void* d_out, int out_size, void* d_ws, size_t ws_size,
                              hipStream_t stream) {
  const float* emb = (const float*)d_in[0];
  const int*   p2d = (const int*)d_in[1];
  // d_in[2] = num_doms scalar (unused; derived from array sizes)
  const float* dmt = (const float*)d_in[3];
  const float* dsc = (const float*)d_in[4];
  const float* dst = (const float*)d_in[5];
  const int*   dpc = (const int*)d_in[6];
  const float* W   = (const float*)d_in[7];
  const float* bv  = (const float*)d_in[8];
  float* out = (float*)d_out;

  int n_pulses = in_sizes[1];
  int n_doms   = in_sizes[3];
  int n_elems  = n_doms * EMBED;

  float* ws_sum = (float*)d_ws;
  float* ws_max = ws_sum + (size_t)n_elems;
  float* ws_cnt = ws_max + (size_t)n_elems;

  // 1) init accumulators (sum=0, max=-FLT_MAX, cnt=0)
  {
    int t = (n_elems + 3) / 4;               // also covers cnt since t >= 32*n_doms
    int g = (t + 255) / 256;
    init_ws<<<g, 256, 0, stream>>>(ws_sum, ws_max, ws_cnt, n_elems, n_doms);
  }
  // 2) scatter: one wave32 per pulse
  {
    long long threads = (long long)n_pulses * 32;
    int g = (int)((threads + 255) / 256);
    scatter_pool<<<g, 256, 0, stream>>>(emb, p2d, ws_sum, ws_max, ws_cnt, n_pulses);
  }
  // 3) finalize + WMMA GEMM: 16 DOM rows per block
  {
    int g = (n_doms + 15) / 16;
    pool_gemm<<<g, 256, 0, stream>>>(ws_sum, ws_max, ws_cnt,
                                     dmt, dsc, dst, dpc, W, bv, out, n_doms);
  }
  (void)n_in; (void)out_size; (void)ws_size;
}